// Modified_MAML_43903155699813
// MI455X (gfx1250) — compile-verified
//
#include <hip/hip_runtime.h>
#include <cstddef>

#define LEN_C 512
#define R_    4
#define S_    8
#define E_    8
#define T_    2048
#define RS_   (R_*S_)              // 32
#define K_TOT (RS_*LEN_C)          // 16384
#define LR_   0.001f
#define GAMMA_ 0.99f

typedef __attribute__((ext_vector_type(2))) float v2f;
typedef __attribute__((ext_vector_type(8))) float v8f;

// ---------------------------------------------------------------------------
// Kernel A: error[e] = Dis[e, LEN_C-1] - sum_{rs,n} Fx[rs,e,n] * W0[rs,n]
// Tiny (131K MACs): one 256-thread block, LDS tree reduce per e.
// ---------------------------------------------------------------------------
__global__ void __launch_bounds__(256) err_kernel(const float* __restrict__ Fx,
                                                  const float* __restrict__ Dis,
                                                  const float* __restrict__ W0,
                                                  float* __restrict__ err) {
    __shared__ float red[256];
    const int tid = threadIdx.x;
    for (int e = 0; e < E_; ++e) {
        float p = 0.f;
        for (int k = tid; k < K_TOT; k += 256) {
            const int rs = k >> 9;
            const int n  = k & (LEN_C - 1);
            p += Fx[rs * (E_ * LEN_C) + e * LEN_C + n] * W0[k];
        }
        red[tid] = p;
        __syncthreads();
        for (int off = 128; off > 0; off >>= 1) {
            if (tid < off) red[tid] += red[tid + off];
            __syncthreads();
        }
        if (tid == 0) err[e] = Dis[e * LEN_C + (LEN_C - 1)] - red[0];
        __syncthreads();
    }
}

// ---------------------------------------------------------------------------
// Kernel B: cw[rs,n] = W0[rs,n] + LR * sum_e err[e]*Fx[rs,e,n]
//           (== W0 - 0.5*LR * (-2 * sum_e err*Fx), the reference grad step)
// Also emits gam_vector[i] = GAMMA^(LEN_C-1-i) at the tail of d_out.
// Grid: 64 x 256 = 16384 threads (one per weight tap).
// ---------------------------------------------------------------------------
__global__ void __launch_bounds__(256) cw_kernel(const float* __restrict__ Fx,
                                                 const float* __restrict__ W0,
                                                 const float* __restrict__ err,
                                                 float* __restrict__ cw,
                                                 float* __restrict__ gam) {
    const int k  = blockIdx.x * 256 + threadIdx.x;
    const int rs = k >> 9;
    const int n  = k & (LEN_C - 1);
    float g = 0.f;
#pragma unroll
    for (int e = 0; e < E_; ++e)
        g += err[e] * Fx[rs * (E_ * LEN_C) + e * LEN_C + n];
    cw[k] = W0[k] + LR_ * g;
    if (k < LEN_C)
        gam[k] = powf(GAMMA_, (float)(LEN_C - 1 - k));
}

// ---------------------------------------------------------------------------
// Kernel C (bandwidth-bound main pass, 1 GiB streamed once):
//   anti[t,e] = sum_{rs,n} Fx_extend[t,rs,e,n] * cw[rs,n]
// WMMA mapping (V_WMMA_F32_16X16X4_F32, wave32):
//   - D tile rows M = 16 (t,e) pairs: t = 2*blockIdx + (m>>3), e = m&7.
//   - A layout per ISA: lanes 0-15 hold (M=lane, K=0..1), lanes 16-31 hold
//     (M=lane-16, K=2..3) -> each lane does one contiguous float2 load from
//     its (t,e) row stream of Fx_extend.
//   - B broadcasts cw[k] across all 16 N columns, so every D column equals
//     the desired dot product; we extract column N=0 (lanes 0 and 16).
//   - 8 waves/block each reduce a 2048-wide K slice (4 rs rows x 512 taps),
//     then cross-wave reduce through LDS.
// cw (64 KB) is staged in LDS once per block and the buffer is reused for the
// 8x16 partial-sum exchange after a barrier.
// ---------------------------------------------------------------------------
__global__ void __launch_bounds__(256) anti_kernel(const float* __restrict__ Fxe,
                                                   const float* __restrict__ cw,
                                                   float* __restrict__ out) {
    __shared__ float cwlds[K_TOT];   // 64 KB
    const int tid = threadIdx.x;

    // Stage cw -> LDS (float4 vectorized).
    {
        const float4* s  = (const float4*)cw;
        float4*       d4 = (float4*)cwlds;
        for (int i = tid; i < K_TOT / 4; i += 256) d4[i] = s[i];
    }
    __syncthreads();

    const int wave = tid >> 5;
    const int lane = tid & 31;
    const int hi   = lane >> 4;            // 0: K=0,1   1: K=2,3
    const int m    = lane & 15;            // D-tile row
    const int t    = (blockIdx.x << 1) + (m >> 3);
    const int e    = m & 7;

    v8f acc = {};
    for (int rsb = 0; rsb < 4; ++rsb) {
        const int rs = (wave << 2) + rsb;
        const float* __restrict__ rowp =
            Fxe + ((size_t)t * RS_ + rs) * (E_ * LEN_C) + e * LEN_C + (hi << 1);
        // Read ahead: this lane's next rs-row stream (speculative prefetch,
        // OOB translations are silently dropped per ISA).
        __builtin_prefetch(rowp + E_ * LEN_C, 0, 1);
        const int kb0 = (rs << 9) + hi;
#pragma unroll 4
        for (int nb = 0; nb < LEN_C; nb += 4) {
            const v2f a = *(const v2f*)rowp;          // A: K pair for this lane
            v2f b;                                     // B: cw broadcast over N
            b.x = cwlds[kb0 + nb];
            b.y = cwlds[kb0 + nb + 2];
            acc = __builtin_amdgcn_wmma_f32_16x16x4_f32(
                /*neg_a=*/false, a, /*neg_b=*/false, b,
                /*c_mod=*/(short)0, acc, /*reuse_a=*/false, /*reuse_b=*/false);
            rowp += 4;
        }
    }

    // All waves done reading cwlds -> safe to reuse it for partial sums.
    __syncthreads();
    float* part = cwlds;                  // [8 waves][16 rows]
    if (lane == 0) {
#pragma unroll
        for (int j = 0; j < 8; ++j) part[wave * 16 + j] = acc[j];       // M=j
    }
    if (lane == 16) {
#pragma unroll
        for (int j = 0; j < 8; ++j) part[wave * 16 + 8 + j] = acc[j];   // M=j+8
    }
    __syncthreads();
    if (tid < 16) {
        float s = 0.f;
#pragma unroll
        for (int w = 0; w < 8; ++w) s += part[w * 16 + tid];
        // out flat index = (t0 + m>>3)*E + (m&7) = blockIdx*16 + m : contiguous
        out[blockIdx.x * 16 + tid] = s;
    }
}

// ---------------------------------------------------------------------------
extern "C" void kernel_launch(void* const* d_in, const int* in_sizes, int n_in,
                              void* d_out, int out_size, void* d_ws, size_t ws_size,
                              hipStream_t stream) {
    const float* Fx  = (const float*)d_in[0];   // [R,S,E,LEN_C]
    const float* Dis = (const float*)d_in[1];   // [E,LEN_C]
    const float* Fxe = (const float*)d_in[2];   // [T,R,S,E,LEN_C]
    const float* W0  = (const float*)d_in[3];   // [R,S,LEN_C]
    (void)in_sizes; (void)n_in; (void)out_size; (void)ws_size;

    float* ws  = (float*)d_ws;
    float* err = ws;            // 8 floats
    float* cw  = ws + 8;        // 16384 floats (32B-offset -> 16B aligned)

    float* out = (float*)d_out;            // [T*E] anti_noise
    float* gam = out + (size_t)T_ * E_;    // [LEN_C] gamma vector

    err_kernel<<<1, 256, 0, stream>>>(Fx, Dis, W0, err);
    cw_kernel<<<K_TOT / 256, 256, 0, stream>>>(Fx, W0, err, cw, gam);
    anti_kernel<<<T_ / 2, 256, 0, stream>>>(Fxe, cw, out);
}